// FusedMHCLayer_32633161515420
// MI455X (gfx1250) — compile-verified
//
#include <hip/hip_runtime.h>
#include <hip/hip_bf16.h>
#include <stdint.h>

#define NN 4
#define TMAXI 20
#define EPSF 1e-6f
#define POST_MULTF 2.0f
#define C_DIM 1024
#define C4 (C_DIM / 4) // 256 float4 chunks per 1024-channel row

typedef float v4f __attribute__((ext_vector_type(4)));

// ---------------------------------------------------------------------------
// CDNA5 async global->LDS copy (ASYNCcnt-tracked) + wait, via inline asm.
// vdst = LDS byte address (VGPR), vaddr = 64-bit global address (VGPR pair),
// saddr = off (GV mode). NT hint: streamed data, read exactly once.
// ---------------------------------------------------------------------------
__device__ __forceinline__ void async_copy_b128_to_lds(uint32_t lds_byte_addr,
                                                       unsigned long long gaddr) {
  asm volatile("global_load_async_to_lds_b128 %0, %1, off th:TH_LOAD_NT"
               :
               : "v"(lds_byte_addr), "v"(gaddr)
               : "memory");
}

__device__ __forceinline__ void wait_async0() {
  asm volatile("s_wait_asynccnt 0" ::: "memory");
}

__device__ __forceinline__ uint32_t lds_offset_of(const void* p) {
  // Generic shared pointers on gfx1250 are {shared_aperture, lds_offset[31:0]}.
  return (uint32_t)(uintptr_t)p;
}

// ---------------------------------------------------------------------------
// Kernel B (defined first so it leads the disassembly): bandwidth kernel.
// One block per token. Async-stage the 20 KB tile (4 x-streams + f_out row)
// into LDS, then each lane produces 5 float4 outputs (y_pre + 4 remixed
// streams) with coalesced non-temporal b128 stores.
// ---------------------------------------------------------------------------
__global__ void __launch_bounds__(256)
mhc_stream_kernel(const float* __restrict__ x,
                  const float* __restrict__ f_out,
                  const float* __restrict__ coef,
                  float* __restrict__ y_pre,
                  float* __restrict__ out) {
  const int t = blockIdx.x;
  const int tid = threadIdx.x; // 0..255, one float4 chunk of C per lane

  __shared__ v4f s_tile[5 * C4]; // rows 0..3 = x streams, row 4 = f_out
  __shared__ float s_coef[24];

  const v4f* xg = (const v4f*)(x + (size_t)t * NN * C_DIM);
  const v4f* fg = (const v4f*)(f_out + (size_t)t * C_DIM);

  // Issue 5 async b128 copies per lane (ASYNCcnt path).
#pragma unroll
  for (int n = 0; n < NN; ++n) {
    async_copy_b128_to_lds(lds_offset_of(&s_tile[n * C4 + tid]),
                           (unsigned long long)(uintptr_t)(xg + n * C4 + tid));
  }
  async_copy_b128_to_lds(lds_offset_of(&s_tile[4 * C4 + tid]),
                         (unsigned long long)(uintptr_t)(fg + tid));

  if (tid < 24) s_coef[tid] = coef[(size_t)t * 24 + tid];
  __syncthreads();   // publish coefficients
  wait_async0();     // each lane waits for its own async copies

  float hp[NN], hq[NN], R[NN][NN];
#pragma unroll
  for (int n = 0; n < NN; ++n) {
    hp[n] = s_coef[n];
    hq[n] = s_coef[NN + n];
  }
#pragma unroll
  for (int j = 0; j < NN; ++j)
#pragma unroll
    for (int i = 0; i < NN; ++i) R[j][i] = s_coef[2 * NN + j * NN + i];

  v4f xv[NN];
#pragma unroll
  for (int n = 0; n < NN; ++n) xv[n] = s_tile[n * C4 + tid];
  v4f fv = s_tile[4 * C4 + tid];

  // y_pre = sum_n h_pre[n] * x[n, :]
  v4f y = hp[0] * xv[0] + hp[1] * xv[1] + hp[2] * xv[2] + hp[3] * xv[3];
  __builtin_nontemporal_store(y, (v4f*)(y_pre + (size_t)t * C_DIM) + tid);

  // out[j, :] = sum_i h_res[j,i] * x[i, :] + h_post[j] * f_out[:]
#pragma unroll
  for (int j = 0; j < NN; ++j) {
    v4f o = hq[j] * fv;
#pragma unroll
    for (int i = 0; i < NN; ++i) o += R[j][i] * xv[i];
    __builtin_nontemporal_store(
        o, (v4f*)(out + ((size_t)t * NN + j) * C_DIM) + tid);
  }
}

// ---------------------------------------------------------------------------
// Kernel A: per-token coefficients (sigmoid gates + Sinkhorn 4x4), registers
// only. Output layout per token: [0..3]=h_pre, [4..7]=h_post, [8..23]=h_res
// row-major (j,i). Written with default (RT) policy: reused by kernel B via L2.
// ---------------------------------------------------------------------------
__global__ void mhc_coef_kernel(const float* __restrict__ mix,
                                const float* __restrict__ invr,
                                const float* __restrict__ alpha_pre,
                                const float* __restrict__ alpha_post,
                                const float* __restrict__ alpha_res,
                                const float* __restrict__ bias,
                                float* __restrict__ coef, int T) {
  int t = blockIdx.x * blockDim.x + threadIdx.x;
  if (t >= T) return;

  const float inv = invr[t];
  const float ap = alpha_pre[0];
  const float aq = alpha_post[0];
  const float ar = alpha_res[0];

  float out24[24];

#pragma unroll
  for (int n = 0; n < NN; ++n) {
    float sp = mix[t * 24 + n] * inv;
    float sq = mix[t * 24 + NN + n] * inv;
    out24[n]      = 1.0f / (1.0f + __expf(-(fmaf(sp, ap, bias[n]))));
    out24[NN + n] = POST_MULTF / (1.0f + __expf(-(fmaf(sq, aq, bias[NN + n]))));
  }

  // Sinkhorn: row-softmax exp, then TMAX row/col normalizations.
  float r[NN][NN];
#pragma unroll
  for (int j = 0; j < NN; ++j) {
    float z[NN];
    float m = -INFINITY;
#pragma unroll
    for (int i = 0; i < NN; ++i) {
      int k = 2 * NN + j * NN + i;
      z[i] = fmaf(mix[t * 24 + k] * inv, ar, bias[k]);
      m = fmaxf(m, z[i]);
    }
#pragma unroll
    for (int i = 0; i < NN; ++i) r[j][i] = __expf(z[i] - m);
  }

  for (int it = 0; it < TMAXI; ++it) {
#pragma unroll
    for (int j = 0; j < NN; ++j) {
      float rs = r[j][0] + r[j][1] + r[j][2] + r[j][3] + EPSF;
      float ri = 1.0f / rs;
#pragma unroll
      for (int i = 0; i < NN; ++i) r[j][i] *= ri;
    }
#pragma unroll
    for (int i = 0; i < NN; ++i) {
      float cs = r[0][i] + r[1][i] + r[2][i] + r[3][i] + EPSF;
      float ci = 1.0f / cs;
#pragma unroll
      for (int j = 0; j < NN; ++j) r[j][i] *= ci;
    }
  }

#pragma unroll
  for (int j = 0; j < NN; ++j)
#pragma unroll
    for (int i = 0; i < NN; ++i) out24[2 * NN + j * NN + i] = r[j][i];

#pragma unroll
  for (int k = 0; k < 24; ++k) coef[t * 24 + k] = out24[k];
}

// ---------------------------------------------------------------------------
// Host launcher
// ---------------------------------------------------------------------------
extern "C" void kernel_launch(void* const* d_in, const int* in_sizes, int n_in,
                              void* d_out, int out_size, void* d_ws, size_t ws_size,
                              hipStream_t stream) {
  const float* x          = (const float*)d_in[0]; // (T, N, C)
  const float* f_out      = (const float*)d_in[1]; // (T, C)
  const float* mix        = (const float*)d_in[2]; // (T, 24)
  const float* invr       = (const float*)d_in[3]; // (T,)
  const float* alpha_pre  = (const float*)d_in[4];
  const float* alpha_post = (const float*)d_in[5];
  const float* alpha_res  = (const float*)d_in[6];
  const float* bias       = (const float*)d_in[7]; // (24,)

  const int T = in_sizes[3]; // invr length

  float* coef = (float*)d_ws;                        // T * 24 floats
  float* y_pre = (float*)d_out;                      // (T, C)
  float* out = (float*)d_out + (size_t)T * C_DIM;    // (T, N, C)

  mhc_coef_kernel<<<(T + 127) / 128, 128, 0, stream>>>(
      mix, invr, alpha_pre, alpha_post, alpha_res, bias, coef, T);

  mhc_stream_kernel<<<T, 256, 0, stream>>>(x, f_out, coef, y_pre, out);
}